// Glob_89000312308387
// MI455X (gfx1250) — compile-verified
//
#include <hip/hip_runtime.h>
#include <math.h>

#define NN   4096
#define EE   16384
#define PHID 64
#define EPSV 1e-8f
#define BETA 0.5f

typedef float v2f __attribute__((ext_vector_type(2)));
typedef float v8f __attribute__((ext_vector_type(8)));

// ---------------------------------------------- fm = 0.5*(f0+f1), fm2 = fm*fm
__global__ void fm_kernel(const float* __restrict__ f0, const float* __restrict__ f1,
                          float* __restrict__ fm, float* __restrict__ fm2) {
    int i = blockIdx.x * blockDim.x + threadIdx.x;
    if (i < NN) {
        float v = 0.5f * (f0[i] + f1[i]);
        fm[i]  = v;
        fm2[i] = v * v;
    }
}

// ------------------------------------------- per-node norms: ||A[r,:]*fm + eps||_2
// one wave32 per row; 8 waves per block
__global__ __launch_bounds__(256) void norm_kernel(const float* __restrict__ A,
                                                   const float* __restrict__ fm,
                                                   float* __restrict__ nrm) {
    int wave = threadIdx.x >> 5, lane = threadIdx.x & 31;
    int row  = blockIdx.x * 8 + wave;
    const float* ar = A + (size_t)row * NN;
    float s = 0.f;
    for (int k = lane; k < NN; k += 32) {
        float v = fmaf(ar[k], fm[k], EPSV);
        s = fmaf(v, v, s);
    }
    for (int off = 16; off; off >>= 1) s += __shfl_xor(s, off, 32);
    if (lane == 0) nrm[row] = sqrtf(s);
}

// ------------------------------------------------------------------- edge kernel
// grid = (E/(16*8), 2); one wave per 16-edge tile; layer = blockIdx.y
__global__ __launch_bounds__(256) void edge_kernel(
    const float* __restrict__ A,
    const int*  __restrict__ src0, const int* __restrict__ dst0,
    const int*  __restrict__ src1, const int* __restrict__ dst1,
    const float* __restrict__ fm2, const float* __restrict__ nrm,
    const float* __restrict__ p1w1, const float* __restrict__ p1b1,
    const float* __restrict__ p1w2, const float* __restrict__ p1b2,
    const float* __restrict__ p2w1, const float* __restrict__ p2b1,
    const float* __restrict__ p2w2, const float* __restrict__ p2b2,
    float* __restrict__ out)
{
    __shared__ float fmS[NN];        // 16 KB: fm^2 staged once per block
    __shared__ float diagS[8 * 256]; // 8 KB: per-wave C-tile dump

    const int lid = blockIdx.y;
    const int* __restrict__ src = lid ? src1 : src0;
    const int* __restrict__ dst = lid ? dst1 : dst0;

    for (int k = threadIdx.x; k < NN; k += 256) fmS[k] = fm2[k];
    __syncthreads();

    const int wave = threadIdx.x >> 5, lane = threadIdx.x & 31;
    const int tile = blockIdx.x * 8 + wave;     // 1024 tiles per layer
    const int m    = lane & 15;                 // edge-in-tile (M for A, N for B)
    const int e    = tile * 16 + m;

    const int sIdx = src[e], dIdx = dst[e];
    const float* __restrict__ aS = A + (size_t)sIdx * NN;
    const float* __restrict__ aD = A + (size_t)dIdx * NN;
    const int khalf = (lane >> 4) * 2;          // lanes 0-15: K0,K1; 16-31: K2,K3

    // s1 = sum_k (A[s,k]*fm^2[k]) * A[d,k]  -- scale folded into A operand only
    v8f acc = {};
    #pragma unroll 4
    for (int k0 = 0; k0 < NN; k0 += 4) {
        int k = k0 + khalf;
        v2f fv = *(const v2f*)(&fmS[k]);
        v2f av = *(const v2f*)(aS + k);
        v2f bv = *(const v2f*)(aD + k);
        v2f a; a.x = av.x * fv.x; a.y = av.y * fv.y;   // A: 16 src rows x K (pre-scaled)
        acc = __builtin_amdgcn_wmma_f32_16x16x4_f32(
            /*neg_a=*/false, a, /*neg_b=*/false, bv,
            /*c_mod=*/(short)0, acc, /*reuse_a=*/false, /*reuse_b=*/false);
    }

    // dump C tile, pull diagonal (16x16 f32 C/D layout), branchless index:
    //   m<8 : VGPR m,   lane m     -> m*33
    //   m>=8: VGPR m-8, lane m+16  -> (m-8)*32 + m + 16
    float* dslot = &diagS[wave * 256];
    #pragma unroll
    for (int v = 0; v < 8; ++v) dslot[v * 32 + lane] = acc[v];
    __syncthreads();

    if (lane < 16) {
        int didx = (m & 7) * 32 + m + ((m >> 3) << 4);
        float s1 = dslot[didx];
        float xn = s1 / (nrm[sIdx] * nrm[dIdx]);

        // MLP1: 1 -> 64 -> 1
        const float* W1 = p1w1 + lid * PHID;
        const float* B1 = p1b1 + lid * PHID;
        const float* W2 = p1w2 + lid * PHID;
        float y1 = p1b2[lid];
        #pragma unroll 8
        for (int p = 0; p < PHID; ++p) {
            float h = fmaf(xn, W1[p], B1[p]);
            h = h > 0.f ? h : 0.f;
            y1 = fmaf(h, W2[p], y1);
        }
        float x2 = BETA * y1;   // / LM (== 1)

        // MLP2: 1 -> 64 -> 1
        const float* V1 = p2w1 + lid * PHID;
        const float* C1 = p2b1 + lid * PHID;
        const float* V2 = p2w2 + lid * PHID;
        float y2 = p2b2[lid];
        #pragma unroll 8
        for (int p = 0; p < PHID; ++p) {
            float h = fmaf(x2, V1[p], C1[p]);
            h = h > 0.f ? h : 0.f;
            y2 = fmaf(h, V2[p], y2);
        }
        out[lid * EE + e] = y2;
    }
}

extern "C" void kernel_launch(void* const* d_in, const int* in_sizes, int n_in,
                              void* d_out, int out_size, void* d_ws, size_t ws_size,
                              hipStream_t stream) {
    const float* A    = (const float*)d_in[0];
    const int*   src0 = (const int*)  d_in[1];
    const int*   dst0 = (const int*)  d_in[2];
    const int*   src1 = (const int*)  d_in[3];
    const int*   dst1 = (const int*)  d_in[4];
    const float* f0   = (const float*)d_in[5];
    const float* f1   = (const float*)d_in[6];
    const float* p1w1 = (const float*)d_in[7];
    const float* p1b1 = (const float*)d_in[8];
    const float* p1w2 = (const float*)d_in[9];
    const float* p1b2 = (const float*)d_in[10];
    const float* p2w1 = (const float*)d_in[11];
    const float* p2b1 = (const float*)d_in[12];
    const float* p2w2 = (const float*)d_in[13];
    const float* p2b2 = (const float*)d_in[14];

    float* fm  = (float*)d_ws;   // N floats
    float* fm2 = fm  + NN;       // N floats
    float* nrm = fm2 + NN;       // N floats (48 KB total scratch)

    fm_kernel  <<<NN / 256, 256, 0, stream>>>(f0, f1, fm, fm2);
    norm_kernel<<<NN / 8,   256, 0, stream>>>(A, fm, nrm);

    dim3 grid(EE / (16 * 8), 2);  // (128, 2)
    edge_kernel<<<grid, 256, 0, stream>>>(A, src0, dst0, src1, dst1, fm2, nrm,
                                          p1w1, p1b1, p1w2, p1b2,
                                          p2w1, p2b1, p2w2, p2b2,
                                          (float*)d_out);
}